// SAGE_18382460027034
// MI455X (gfx1250) — compile-verified
//
#include <hip/hip_runtime.h>
#include <hip/hip_bf16.h>

typedef float v2f __attribute__((ext_vector_type(2)));
typedef float v8f __attribute__((ext_vector_type(8)));

#define D_FEAT 128

// ---------------------------------------------------------------------------
// Hardware fp32 atomic add (no return) -> global_atomic_add_f32, STOREcnt
// ---------------------------------------------------------------------------
__device__ __forceinline__ void atomic_add_f32(float* p, float v) {
    asm volatile("global_atomic_add_f32 %0, %1, off" :: "v"(p), "v"(v) : "memory");
}

// ---------------------------------------------------------------------------
// Zero-fill (grid stride)
// ---------------------------------------------------------------------------
__global__ void zero_f32(float* __restrict__ p, long n) {
    long i = (long)blockIdx.x * blockDim.x + threadIdx.x;
    long stride = (long)gridDim.x * blockDim.x;
    for (; i < n; i += stride) p[i] = 0.0f;
}

// ---------------------------------------------------------------------------
// In-degree via fp32 atomics
// ---------------------------------------------------------------------------
__global__ void degree_k(const int* __restrict__ dst, float* __restrict__ deg, int E) {
    int e = blockIdx.x * blockDim.x + threadIdx.x;
    if (e < E) atomic_add_f32(&deg[dst[e]], 1.0f);
}

// deg -> 1 / max(deg, 1)  (in place)
__global__ void invdeg_k(float* __restrict__ deg, int N) {
    int i = blockIdx.x * blockDim.x + threadIdx.x;
    if (i < N) {
        float d = deg[i];
        deg[i] = 1.0f / fmaxf(d, 1.0f);
    }
}

// ---------------------------------------------------------------------------
// Edge scatter: agg[dst] += feat[src].  One wave per edge: 32 lanes x float4
// covers the 128-feature row.  Coalesced b128 gather, 4 fp32 atomics per lane.
// ---------------------------------------------------------------------------
__global__ void scatter_k(const float* __restrict__ feat,
                          const int* __restrict__ src,
                          const int* __restrict__ dst,
                          float* __restrict__ agg, int E) {
    long idx = (long)blockIdx.x * blockDim.x + threadIdx.x;
    if (idx >= (long)E * 32) return;
    int e = (int)(idx >> 5);
    int c = (int)(idx & 31) * 4;
    int s = src[e];
    int t = dst[e];
    const float4 v = *(const float4*)(feat + (size_t)s * D_FEAT + c);
    float* a = agg + (size_t)t * D_FEAT + c;
    atomic_add_f32(a + 0, v.x);
    atomic_add_f32(a + 1, v.y);
    atomic_add_f32(a + 2, v.z);
    atomic_add_f32(a + 3, v.w);
}

// ---------------------------------------------------------------------------
// Fused SAGE GEMM:  Out = act( Xs @ Ws + (invdeg .* Agg) @ Wn + b )
//   K = 128, fp32 WMMA (V_WMMA_F32_16X16X4_F32).
//   Block = 128 threads (4 waves).  Wave w -> 16-row tile; block covers 64 rows.
//   blockIdx.y selects a 64-column slab (DOUT=128 -> 2 slabs, DOUT=64 -> 1).
//   Weight slabs staged in LDS, swizzled so a B fragment is one ds_load_b64:
//     LDS float index of W[k][n]  =  ((k>>2)*64 + n)*4 + (k&3)
//   A/B fragment lane layout (ISA 7.12.2, 32-bit 16x4):
//     lanes 0-15 : (K=0,K=1) for M/N = lane ;  lanes 16-31 : (K=2,K=3)
// ---------------------------------------------------------------------------
template <int DOUT, bool RELU>
__global__ __launch_bounds__(128) void sage_gemm(
    const float* __restrict__ Xs,     // [N,128] self features
    const float* __restrict__ Agg,    // [N,128] summed neighbor features
    const float* __restrict__ invdeg, // [N]     1/max(deg,1)
    const float* __restrict__ Ws,     // [128,DOUT]
    const float* __restrict__ Wn,     // [128,DOUT]
    const float* __restrict__ bias,   // [DOUT]
    float* __restrict__ Out,          // [N,DOUT]
    int N)
{
    __shared__ float sWs[D_FEAT * 64];
    __shared__ float sWn[D_FEAT * 64];

    const int n0 = blockIdx.y * 64;

    // cooperative, swizzled stage of both 128x64 weight slabs
    for (int j = threadIdx.x; j < D_FEAT * 64; j += 128) {
        int k = j >> 6;
        int n = j & 63;
        int sidx = ((k >> 2) * 64 + n) * 4 + (k & 3);
        sWs[sidx] = Ws[k * DOUT + n0 + n];
        sWn[sidx] = Wn[k * DOUT + n0 + n];
    }
    __syncthreads();

    const int lane = threadIdx.x & 31;
    const int wave = threadIdx.x >> 5;
    const int row0 = (blockIdx.x * 4 + wave) * 16;
    if (row0 >= N) return;                 // wave-uniform; EXEC stays all-ones

    const int m    = lane & 15;            // row (A) / col (B,C) within tile
    const int koff = (lane >> 4) * 2;      // K sub-offset for this half-wave

    const float invd = invdeg[row0 + m];
    const float* xrow = Xs  + (size_t)(row0 + m) * D_FEAT + koff;
    const float* arow = Agg + (size_t)(row0 + m) * D_FEAT + koff;

    v8f acc[4];
#pragma unroll
    for (int j = 0; j < 4; ++j) {
        float bv = bias[n0 + j * 16 + m];  // same column -> same bias in all 8 rows
#pragma unroll
        for (int t = 0; t < 8; ++t) acc[j][t] = bv;
    }

#pragma unroll 4
    for (int g = 0; g < 32; ++g) {         // K = 128 in steps of 4
        v2f ax = *(const v2f*)(xrow + 4 * g);
        v2f ah = *(const v2f*)(arow + 4 * g);
        ah.x *= invd;                      // fold mean normalization into A
        ah.y *= invd;
        const int bbase = (g * 64 + m) * 4 + koff;
#pragma unroll
        for (int j = 0; j < 4; ++j) {
            v2f bs = *(const v2f*)&sWs[bbase + j * 64];
            acc[j] = __builtin_amdgcn_wmma_f32_16x16x4_f32(
                false, ax, false, bs, (short)0, acc[j], false, false);
            v2f bn = *(const v2f*)&sWn[bbase + j * 64];
            acc[j] = __builtin_amdgcn_wmma_f32_16x16x4_f32(
                false, ah, false, bn, (short)0, acc[j], false, false);
        }
    }

    // C/D layout: vgpr r, lanes 0-15 -> row r, lanes 16-31 -> row r+8; col = lane&15
    const int rbase = row0 + (lane >> 4) * 8;
#pragma unroll
    for (int j = 0; j < 4; ++j) {
        int col = n0 + j * 16 + m;
#pragma unroll
        for (int r = 0; r < 8; ++r) {
            float v = acc[j][r];
            if (RELU) v = fmaxf(v, 0.0f);
            Out[(size_t)(rbase + r) * DOUT + col] = v;
        }
    }
}

// ---------------------------------------------------------------------------
// Launch
// ---------------------------------------------------------------------------
extern "C" void kernel_launch(void* const* d_in, const int* in_sizes, int n_in,
                              void* d_out, int out_size, void* d_ws, size_t ws_size,
                              hipStream_t stream) {
    const float* x    = (const float*)d_in[0];
    const int*   esrc = (const int*)d_in[1];
    const int*   edst = (const int*)d_in[2];
    const float* Ws0  = (const float*)d_in[3];
    const float* Wn0  = (const float*)d_in[4];
    const float* b0   = (const float*)d_in[5];
    const float* Ws1  = (const float*)d_in[6];
    const float* Wn1  = (const float*)d_in[7];
    const float* b1   = (const float*)d_in[8];
    const float* Ws2  = (const float*)d_in[9];
    const float* Wn2  = (const float*)d_in[10];
    const float* b2   = (const float*)d_in[11];

    const int N = in_sizes[0] / D_FEAT;   // 50000
    const int E = in_sizes[1];            // 800000

    // workspace layout (floats): deg/invdeg | agg | h0 | h1
    float* ws   = (float*)d_ws;
    const size_t degPad = (size_t)((N + 127) / 128) * 128;
    const size_t nd     = (size_t)N * D_FEAT;
    float* deg = ws;
    float* agg = ws + degPad;
    float* h0  = agg + nd;
    float* h1  = h0 + nd;

    const int scatBlocks = (int)(((long)E * 32 + 255) / 256);
    const int gx = ((N + 15) / 16 + 3) / 4;   // 16-row tiles, 4 waves/block

    // graph degree (static across layers)
    zero_f32<<<1024, 256, 0, stream>>>(deg, (long)N);
    degree_k<<<(E + 255) / 256, 256, 0, stream>>>(edst, deg, E);
    invdeg_k<<<(N + 255) / 256, 256, 0, stream>>>(deg, N);

    // ---- layer 0: x -> h0 (relu) ----
    zero_f32<<<4096, 256, 0, stream>>>(agg, (long)nd);
    scatter_k<<<scatBlocks, 256, 0, stream>>>(x, esrc, edst, agg, E);
    sage_gemm<128, true><<<dim3(gx, 2), 128, 0, stream>>>(
        x, agg, deg, Ws0, Wn0, b0, h0, N);

    // ---- layer 1: h0 -> h1 (relu) ----
    zero_f32<<<4096, 256, 0, stream>>>(agg, (long)nd);
    scatter_k<<<scatBlocks, 256, 0, stream>>>(h0, esrc, edst, agg, E);
    sage_gemm<128, true><<<dim3(gx, 2), 128, 0, stream>>>(
        h0, agg, deg, Ws1, Wn1, b1, h1, N);

    // ---- layer 2: h1 -> out (no relu, DOUT=64) ----
    zero_f32<<<4096, 256, 0, stream>>>(agg, (long)nd);
    scatter_k<<<scatBlocks, 256, 0, stream>>>(h1, esrc, edst, agg, E);
    sage_gemm<64, false><<<dim3(gx, 1), 128, 0, stream>>>(
        h1, agg, deg, Ws2, Wn2, b2, (float*)d_out, N);
}